// GAT_36979668418993
// MI455X (gfx1250) — compile-verified
//
#include <hip/hip_runtime.h>
#include <hip/hip_bf16.h>
#include <math.h>

// ---------------------------------------------------------------------------
// GAT (2-layer, PyG-style) for MI455X / gfx1250.
// wave32 + V_WMMA_F32_16X16X4_F32 + async global->LDS double-buffered staging.
// ---------------------------------------------------------------------------

typedef float v2f __attribute__((ext_vector_type(2)));
typedef float v8f __attribute__((ext_vector_type(8)));

#define NEG_SLOPE 0.2f

__device__ __forceinline__ float lrelu(float x) { return x > 0.f ? x : NEG_SLOPE * x; }
__device__ __forceinline__ float eluf(float x)  { return x > 0.f ? x : (__expf(x) - 1.f); }

// Sign-aware float atomic max using integer atomics (IEEE ordering trick).
__device__ __forceinline__ float atomicMaxFloat(float* addr, float v) {
  return (v >= 0.f)
      ? __int_as_float(atomicMax((int*)addr, __float_as_int(v)))
      : __uint_as_float(atomicMin((unsigned int*)addr, __float_as_uint(v)));
}

// --------------------- gfx1250 async global->LDS helpers -------------------
#define HAVE_ASYNC_LDS __has_builtin(__builtin_amdgcn_global_load_async_to_lds_b32)

typedef __attribute__((address_space(1))) int* gint_p;
typedef __attribute__((address_space(3))) int* lint_p;

__device__ __forceinline__ void stage_b32(const float* g, float* l) {
#if HAVE_ASYNC_LDS
  __builtin_amdgcn_global_load_async_to_lds_b32(
      (gint_p)(float*)g, (lint_p)l, 0, 0);
#else
  *l = *g;
#endif
}

__device__ __forceinline__ void async_wait0() {
#if HAVE_ASYNC_LDS
# if __has_builtin(__builtin_amdgcn_s_wait_asynccnt)
  __builtin_amdgcn_s_wait_asynccnt(0);
# else
  asm volatile("s_wait_asynccnt 0x0" ::: "memory");
# endif
#endif
}

// ---------------------------------------------------------------------------
// Fill kernel (workspace init: 0 for accumulators, -inf for segment max)
// ---------------------------------------------------------------------------
__global__ void fill_kernel(float* __restrict__ p, float v, int n) {
  int i = blockIdx.x * blockDim.x + threadIdx.x;
  if (i < n) p[i] = v;
}

// ---------------------------------------------------------------------------
// WMMA fp32 GEMM: C[M,NCOLS] = A[M,K] @ B[K,NCOLS], row-major.
//
// Block = 256 threads = 8 waves; the block spans the full NCOLS width, so a
// 4 x NCOLS slab of B per K-step is staged once into LDS (async, double
// buffered) and shared by all 8 waves. Each wave owns a 16x(16*TN) output
// strip: A regs are reused across TN WMMAs, and A rows are read from global
// once per wave, i.e. GROUPS_N times per kernel.
//
// Wave32 register layouts per CDNA5 ISA 7.12.2:
//   A 16x4 : lanes 0-15 rows 0-15 {K=k,k+1}; lanes 16-31 same rows {K=k+2,k+3}
//   B 4x16 : vgpr0 lanes 0-15 = row k, lanes 16-31 = row k+2; vgpr1 = k+1/k+3
//   C/D    : vgpr v: lanes 0-15 = row v, lanes 16-31 = row v+8
//
// No early returns: barriers must stay convergent, so OOB rows clamp loads
// and predicate stores; EXEC is all-ones at every WMMA.
// ---------------------------------------------------------------------------
template <int NCOLS, int TN>
__global__ void __launch_bounds__(256)
wmma_gemm_f32(const float* __restrict__ A, const float* __restrict__ B,
              float* __restrict__ C, int M, int K) {
  constexpr int GROUPS_N   = NCOLS / (16 * TN);  // waves per M-tile row
  constexpr int MT_PER_BLK = 8 / GROUPS_N;       // M-tiles per block
  constexpr int STAGE      = 4 * NCOLS;          // floats staged per K-step
  constexpr int LD_PER_THR = STAGE / 256;

  __shared__ float ldsB[2][STAGE];

  const int tid  = threadIdx.x;
  const int lane = tid & 31;
  const int wv   = tid >> 5;
  const int tg   = wv % GROUPS_N;
  const int tm   = blockIdx.x * MT_PER_BLK + wv / GROUPS_N;

  const int rowA = tm * 16 + (lane & 15);
  const int rowL = rowA < M ? rowA : M - 1;      // clamped load row
  const int col0 = tg * (16 * TN) + (lane & 15);
  const int koff = (lane >> 4) << 1;             // 0 | 2

  const float* Ap = A + (size_t)rowL * K + koff;
  const int Kmain = K & ~3;

  // Prologue: stage B rows [0,4)
  if (Kmain > 0) {
#pragma unroll
    for (int j = 0; j < LD_PER_THR; ++j) {
      const int e = tid + j * 256;
      stage_b32(&B[(size_t)(e / NCOLS) * NCOLS + (e % NCOLS)], &ldsB[0][e]);
    }
  }

  v8f acc[TN];
#pragma unroll
  for (int t = 0; t < TN; ++t) acc[t] = (v8f){};

  int buf = 0;
  for (int kk = 0; kk < Kmain; kk += 4) {
    async_wait0();
    __syncthreads();  // slab `buf` fully resident in LDS for all waves

    const int nxt = kk + 4;
    if (nxt < Kmain) {
#pragma unroll
      for (int j = 0; j < LD_PER_THR; ++j) {
        const int e = tid + j * 256;
        stage_b32(&B[(size_t)(nxt + e / NCOLS) * NCOLS + (e % NCOLS)],
                  &ldsB[buf ^ 1][e]);
      }
    }
    __builtin_prefetch(Ap + kk + 8, 0, 3);  // global_prefetch_b8

    v2f a;
    a.x = Ap[kk];
    a.y = Ap[kk + 1];
    const float* lb = &ldsB[buf][koff * NCOLS];
#pragma unroll
    for (int t = 0; t < TN; ++t) {
      v2f b;
      b.x = lb[col0 + t * 16];
      b.y = lb[NCOLS + col0 + t * 16];
      acc[t] = __builtin_amdgcn_wmma_f32_16x16x4_f32(
          false, a, false, b, (short)0, acc[t], false, false);
    }
    buf ^= 1;
  }

  // K remainder (<=3): branch-free via clamped indices + 0/1 weights.
  if (Kmain < K) {
    const int kb  = Kmain + koff;
    const float w0 = (kb     < K) ? 1.f : 0.f;
    const float w1 = (kb + 1 < K) ? 1.f : 0.f;
    const int   i0 = (kb     < K) ? kb     : K - 1;
    const int   i1 = (kb + 1 < K) ? kb + 1 : K - 1;
    v2f a;
    a.x = A[(size_t)rowL * K + i0] * w0;
    a.y = A[(size_t)rowL * K + i1] * w1;
#pragma unroll
    for (int t = 0; t < TN; ++t) {
      v2f b;
      b.x = B[(size_t)i0 * NCOLS + col0 + t * 16] * w0;
      b.y = B[(size_t)i1 * NCOLS + col0 + t * 16] * w1;
      acc[t] = __builtin_amdgcn_wmma_f32_16x16x4_f32(
          false, a, false, b, (short)0, acc[t], false, false);
    }
  }

  // Store D (predicated on M only; col0 always in range)
  const int rbase = tm * 16 + ((lane >> 4) << 3);
#pragma unroll
  for (int t = 0; t < TN; ++t) {
#pragma unroll
    for (int v = 0; v < 8; ++v) {
      const int r = rbase + v;
      if (r < M) C[(size_t)r * NCOLS + col0 + t * 16] = acc[t][v];
    }
  }
}

// ---------------------------------------------------------------------------
// Per-(node,head) attention scores: s = <h, a_src>, d = <h, a_dst>
// ---------------------------------------------------------------------------
__global__ void attn_scores(const float* __restrict__ h,
                            const float* __restrict__ a_src,
                            const float* __restrict__ a_dst,
                            float* __restrict__ s, float* __restrict__ d,
                            int N, int H, int C) {
  int idx = blockIdx.x * blockDim.x + threadIdx.x;  // node*H + head
  if (idx >= N * H) return;
  const int head = idx % H;
  const float* hp = h + (size_t)(idx / H) * (H * C) + head * C;
  const float* as = a_src + head * C;
  const float* ad = a_dst + head * C;
  float ss = 0.f, dd = 0.f;
  for (int c = 0; c < C; ++c) {
    const float v = hp[c];
    ss += v * as[c];
    dd += v * ad[c];
  }
  s[idx] = ss;
  d[idx] = dd;
}

// ---------------------------------------------------------------------------
// Edge pass 1: segment max of leaky_relu(s[src]+d[dst]) over dst.
// Edge ids >= E are the appended self-loops (src=dst=id-E).
// ---------------------------------------------------------------------------
__global__ void edge_max_kernel(const int* __restrict__ ei, int E, int N, int H,
                                const float* __restrict__ s,
                                const float* __restrict__ d,
                                float* __restrict__ m) {
  int t = blockIdx.x * blockDim.x + threadIdx.x;
  const int ET = E + N;
  if (t >= ET * H) return;
  const int e = t / H, h = t % H;
  const int src = (e < E) ? ei[e]     : (e - E);
  const int dst = (e < E) ? ei[E + e] : (e - E);
  const float ev = lrelu(s[src * H + h] + d[dst * H + h]);
  atomicMaxFloat(&m[dst * H + h], ev);
}

// ---------------------------------------------------------------------------
// Edge pass 2: denom = segment_sum(exp(e - m[dst]))
// ---------------------------------------------------------------------------
__global__ void edge_expsum_kernel(const int* __restrict__ ei, int E, int N, int H,
                                   const float* __restrict__ s,
                                   const float* __restrict__ d,
                                   const float* __restrict__ m,
                                   float* __restrict__ den) {
  int t = blockIdx.x * blockDim.x + threadIdx.x;
  const int ET = E + N;
  if (t >= ET * H) return;
  const int e = t / H, h = t % H;
  const int src = (e < E) ? ei[e]     : (e - E);
  const int dst = (e < E) ? ei[E + e] : (e - E);
  const float ev = lrelu(s[src * H + h] + d[dst * H + h]);
  atomicAdd(&den[dst * H + h], __expf(ev - m[dst * H + h]));
}

// ---------------------------------------------------------------------------
// Edge pass 3: out[dst] += alpha * h[src], one wave per edge.
// Each lane owns a contiguous chunk of HC/32 channels so the h[src] gather
// merges into b64/b128 loads while the wave still covers one 128B-aligned
// row of h[src]. Scatter via global_atomic_add_f32.
// ---------------------------------------------------------------------------
__global__ void edge_aggregate_kernel(const int* __restrict__ ei, int E, int N,
                                      int H, int C,
                                      const float* __restrict__ h,
                                      const float* __restrict__ s,
                                      const float* __restrict__ d,
                                      const float* __restrict__ m,
                                      const float* __restrict__ den,
                                      float* __restrict__ out) {
  const int lane = threadIdx.x & 31;
  const int e    = (int)((blockIdx.x * blockDim.x + threadIdx.x) >> 5);
  const int ET   = E + N;
  if (e >= ET) return;
  const int src = (e < E) ? ei[e]     : (e - E);
  const int dst = (e < E) ? ei[E + e] : (e - E);
  const int HC  = H * C;
  const int per = HC >> 5;            // 8 (layer 1) or 2 (layer 2)
  const int cb  = lane * per;
  const float* hs = h + (size_t)src * HC + cb;
  float*       od = out + (size_t)dst * HC + cb;
  for (int j = 0; j < per; ++j) {
    const int head    = (cb + j) / C;
    const float ev    = lrelu(s[src * H + head] + d[dst * H + head]);
    const float alpha = __expf(ev - m[dst * H + head]) /
                        (den[dst * H + head] + 1e-16f);
    atomicAdd(&od[j], alpha * hs[j]);
  }
}

// ---------------------------------------------------------------------------
// y = elu(x + bias[c])  (in place)
// ---------------------------------------------------------------------------
__global__ void bias_elu_kernel(float* __restrict__ x, const float* __restrict__ b,
                                int n, int HC) {
  int i = blockIdx.x * blockDim.x + threadIdx.x;
  if (i < n) x[i] = eluf(x[i] + b[i % HC]);
}

// ---------------------------------------------------------------------------
// logits = h @ Wc + bc; out = log_softmax(logits)   (NC == 2)
// ---------------------------------------------------------------------------
__global__ void classifier_kernel(const float* __restrict__ h,
                                  const float* __restrict__ Wc,
                                  const float* __restrict__ bc,
                                  float* __restrict__ out, int N, int C) {
  int n = blockIdx.x * blockDim.x + threadIdx.x;
  if (n >= N) return;
  float l0 = bc[0], l1 = bc[1];
  const float* hp = h + (size_t)n * C;
  for (int k = 0; k < C; ++k) {
    const float v = hp[k];
    l0 += v * Wc[k * 2 + 0];
    l1 += v * Wc[k * 2 + 1];
  }
  const float mx  = fmaxf(l0, l1);
  const float lse = mx + __logf(__expf(l0 - mx) + __expf(l1 - mx));
  out[n * 2 + 0] = l0 - lse;
  out[n * 2 + 1] = l1 - lse;
}

// ---------------------------------------------------------------------------
// Host launcher
// ---------------------------------------------------------------------------
static inline int ceil_div(long long a, long long b) { return (int)((a + b - 1) / b); }

extern "C" void kernel_launch(void* const* d_in, const int* in_sizes, int n_in,
                              void* d_out, int out_size, void* d_ws, size_t ws_size,
                              hipStream_t stream) {
  const float* x      = (const float*)d_in[0];
  const int*   ei     = (const int*)d_in[1];   // [2, E] int32
  const float* W1     = (const float*)d_in[2];
  const float* a_src1 = (const float*)d_in[3];
  const float* a_dst1 = (const float*)d_in[4];
  const float* b1     = (const float*)d_in[5];
  const float* W2     = (const float*)d_in[6];
  const float* a_src2 = (const float*)d_in[7];
  const float* a_dst2 = (const float*)d_in[8];
  const float* b2     = (const float*)d_in[9];
  const float* Wc     = (const float*)d_in[10];
  const float* bc     = (const float*)d_in[11];
  float* out = (float*)d_out;

  // Derive dims (reference: N=100000, F=165, HID=64, H1=4, NC=2)
  const int HC1 = in_sizes[5];                 // H1*HID = 256
  const int F   = in_sizes[2] / HC1;           // 165
  const int N   = in_sizes[0] / F;             // 100000
  const int NC  = in_sizes[11];                // 2
  const int HID = in_sizes[10] / NC;           // 64
  const int H1  = HC1 / HID;                   // 4
  const int E   = in_sizes[1] / 2;             // 1600000
  const int ET  = E + N;                       // with self-loops

  // Workspace layout (floats)
  float* w = (float*)d_ws;
  size_t o = 0;
  float* h1   = w + o; o += (size_t)N * HC1;   // x @ W1
  float* agg1 = w + o; o += (size_t)N * HC1;   // layer-1 aggregation (then ELU)
  float* s1   = w + o; o += (size_t)N * H1;
  float* d1   = w + o; o += (size_t)N * H1;
  float* m1   = w + o; o += (size_t)N * H1;
  float* den1 = w + o; o += (size_t)N * H1;
  float* h2   = w + o; o += (size_t)N * HID;   // agg1 @ W2
  float* agg2 = w + o; o += (size_t)N * HID;   // layer-2 aggregation (then ELU)
  float* s2   = w + o; o += (size_t)N;
  float* d2   = w + o; o += (size_t)N;
  float* m2   = w + o; o += (size_t)N;
  float* den2 = w + o; o += (size_t)N;
  (void)ws_size; (void)n_in; (void)out_size;

  const int BLK = 256;
  const float NEG_INF = -__builtin_huge_valf();
  const int Mtiles = (N + 15) / 16;

  // ---- init accumulators ----
  fill_kernel<<<ceil_div((long long)N * HC1, BLK), BLK, 0, stream>>>(agg1, 0.f, N * HC1);
  fill_kernel<<<ceil_div((long long)N * H1, BLK), BLK, 0, stream>>>(m1, NEG_INF, N * H1);
  fill_kernel<<<ceil_div((long long)N * H1, BLK), BLK, 0, stream>>>(den1, 0.f, N * H1);
  fill_kernel<<<ceil_div((long long)N * HID, BLK), BLK, 0, stream>>>(agg2, 0.f, N * HID);
  fill_kernel<<<ceil_div(N, BLK), BLK, 0, stream>>>(m2, NEG_INF, N);
  fill_kernel<<<ceil_div(N, BLK), BLK, 0, stream>>>(den2, 0.f, N);

  // ---- layer 1 ----
  // NCOLS=256, TN=4 -> 4 wave-groups per M-tile, 2 M-tiles per block.
  wmma_gemm_f32<256, 4><<<ceil_div(Mtiles, 2), BLK, 0, stream>>>(x, W1, h1, N, F);
  attn_scores<<<ceil_div((long long)N * H1, BLK), BLK, 0, stream>>>(
      h1, a_src1, a_dst1, s1, d1, N, H1, HID);
  edge_max_kernel<<<ceil_div((long long)ET * H1, BLK), BLK, 0, stream>>>(
      ei, E, N, H1, s1, d1, m1);
  edge_expsum_kernel<<<ceil_div((long long)ET * H1, BLK), BLK, 0, stream>>>(
      ei, E, N, H1, s1, d1, m1, den1);
  edge_aggregate_kernel<<<ceil_div((long long)ET * 32, BLK), BLK, 0, stream>>>(
      ei, E, N, H1, HID, h1, s1, d1, m1, den1, agg1);
  bias_elu_kernel<<<ceil_div((long long)N * HC1, BLK), BLK, 0, stream>>>(
      agg1, b1, N * HC1, HC1);

  // ---- layer 2 (heads=1, concat=False -> identity over single head) ----
  // NCOLS=64, TN=4 -> 1 wave-group per M-tile, 8 M-tiles per block.
  wmma_gemm_f32<64, 4><<<ceil_div(Mtiles, 8), BLK, 0, stream>>>(agg1, W2, h2, N, HC1);
  attn_scores<<<ceil_div(N, BLK), BLK, 0, stream>>>(
      h2, a_src2, a_dst2, s2, d2, N, 1, HID);
  edge_max_kernel<<<ceil_div((long long)ET, BLK), BLK, 0, stream>>>(
      ei, E, N, 1, s2, d2, m2);
  edge_expsum_kernel<<<ceil_div((long long)ET, BLK), BLK, 0, stream>>>(
      ei, E, N, 1, s2, d2, m2, den2);
  edge_aggregate_kernel<<<ceil_div((long long)ET * 32, BLK), BLK, 0, stream>>>(
      ei, E, N, 1, HID, h2, s2, d2, m2, den2, agg2);
  bias_elu_kernel<<<ceil_div((long long)N * HID, BLK), BLK, 0, stream>>>(
      agg2, b2, N * HID, HID);

  // ---- classifier + log_softmax ----
  classifier_kernel<<<ceil_div(N, BLK), BLK, 0, stream>>>(agg2, Wc, bc, out, N, HID);
}